// RNNDecoder_90993177133832
// MI455X (gfx1250) — compile-verified
//
#include <hip/hip_runtime.h>

typedef float v2f __attribute__((ext_vector_type(2)));
typedef float v4f __attribute__((ext_vector_type(4)));
typedef float v8f __attribute__((ext_vector_type(8)));
typedef int   v4i __attribute__((ext_vector_type(4)));

#define B_SZ 64
#define E_SZ 512
#define H_SZ 1024
#define V_SZ 32000
#define L_SZ 128

#if __has_builtin(__builtin_amdgcn_global_load_async_to_lds_b128) && \
    __has_builtin(__builtin_amdgcn_s_wait_asynccnt)
#define USE_ASYNC_LDS 1
#else
#define USE_ASYNC_LDS 0
#endif

#define AS1 __attribute__((address_space(1)))
#define AS3 __attribute__((address_space(3)))

__device__ __forceinline__ AS1 v4i* gp4(const void* p) {
  return (AS1 v4i*)(uintptr_t)p;   // flat global == addrspace(1) address
}
__device__ __forceinline__ AS3 v4i* lp4(void* p) {
  // generic->LDS: low 32 bits of the flat address are the LDS byte address
  return (AS3 v4i*)(uint32_t)(uintptr_t)p;
}

__device__ __forceinline__ float sigmoidf_(float x) { return 1.0f / (1.0f + __expf(-x)); }

// ---------------------------------------------------------------------------
// M=64 GEMM: C[64,N] = A[64,K] @ op(W) + bias, via v_wmma_f32_16x16x4_f32.
//   TRANSB=false: W is [N,K] row-major, compute A@W^T
//   TRANSB=true : W is [K,N] row-major, compute A@W (transposed in LDS, sync)
//   EPI: 0 = none, 1 = tanh
// Block = 128 threads (4 wave32) computing a 64x64 output tile: wave w owns
// rows 16w..16w+15 with 4 accumulators; one A fragment feeds 4 WMMAs/K-step
// (4x less L2 re-read of the A panel vs a 16-wide tile).
// B (weights, HBM stream) is async double-buffered via ASYNCcnt; A (small,
// L2-resident) is sync single-buffered, its latency hidden behind the async
// weight stream.
// ---------------------------------------------------------------------------
template <int EPI, bool TRANSB>
__global__ __launch_bounds__(128) void gemm64_wmma(
    const float* __restrict__ A, const float* __restrict__ W,
    const float* __restrict__ bias, float* __restrict__ C, int K, int N) {
  __shared__ float sA[64][68];     // stride 68 -> conflict-free ds_load_b64
  __shared__ float sB[2][64][68];  // double-buffered weight tile

  const int tid  = threadIdx.x;
  const int lane = tid & 31;
  const int wv   = tid >> 5;
  const int lm   = lane & 15;   // M (A) / N (B) index within fragment
  const int hl   = lane >> 4;   // K-half selector
  const int n0   = blockIdx.x * 64;

  v8f acc[4];
#pragma unroll
  for (int j = 0; j < 4; ++j) acc[j] = (v8f){0.f,0.f,0.f,0.f,0.f,0.f,0.f,0.f};

  auto stage_A = [&](int kc) {   // A panel chunk [64 rows x 64 k], sync
#pragma unroll
    for (int i = 0; i < 8; ++i) {
      int idx = tid + i * 128;
      int r = idx >> 4, c4 = (idx & 15) << 2;
      *(v4f*)&sA[r][c4] = *(const v4f*)(A + (size_t)r * K + kc + c4);
    }
  };

  auto compute_chunk = [&](int bbuf) {  // 16 K-steps, 4 WMMAs per step
#pragma unroll
    for (int ks = 0; ks < 16; ++ks) {
      v2f a = *(const v2f*)&sA[16 * wv + lm][ks * 4 + 2 * hl];
#pragma unroll
      for (int j = 0; j < 4; ++j) {
        v2f b = *(const v2f*)&sB[bbuf][16 * j + lm][ks * 4 + 2 * hl];
        acc[j] = __builtin_amdgcn_wmma_f32_16x16x4_f32(
            /*neg_a=*/false, a, /*neg_b=*/false, b,
            /*c_mod=*/(short)0, acc[j], /*reuse_a=*/false, /*reuse_b=*/false);
      }
    }
  };

  if constexpr (!TRANSB) {
#if USE_ASYNC_LDS
    // 8 per-thread B128 async copies stage the [64n x 64k] weight tile.
    auto issue_B = [&](int kc, int buf) {
#pragma unroll
      for (int i = 0; i < 8; ++i) {
        int idx = tid + i * 128;
        int r = idx >> 4, c4 = (idx & 15) << 2;
        __builtin_amdgcn_global_load_async_to_lds_b128(
            gp4(W + (size_t)(n0 + r) * K + kc + c4), lp4(&sB[buf][r][c4]),
            /*offset=*/0, /*cpol=*/1);        // weights read once -> TH_NT
      }
    };

    const int nk = K >> 6;
    issue_B(0, 0);
    for (int ic = 0; ic < nk; ++ic) {
      const int cur = ic & 1;
      if (ic + 1 < nk) {
        issue_B((ic + 1) << 6, cur ^ 1);  // next chunk streams during compute
        stage_A(ic << 6);                 // L2 latency overlaps async stream
        __builtin_amdgcn_s_wait_asynccnt(8);  // chunk ic done; ic+1 in flight
      } else {
        stage_A(ic << 6);
        __builtin_amdgcn_s_wait_asynccnt(0);
      }
      __syncthreads();          // all staging visible to all waves
      compute_chunk(cur);
      __syncthreads();          // all reads done before buffer reuse
    }
#else
    // Synchronous fallback (register-staged)
    for (int kc = 0; kc < K; kc += 64) {
      stage_A(kc);
#pragma unroll
      for (int i = 0; i < 8; ++i) {
        int idx = tid + i * 128;
        int r = idx >> 4, c4 = (idx & 15) << 2;
        v4f v = __builtin_nontemporal_load(
            (const v4f*)(W + (size_t)(n0 + r) * K + kc + c4));
        *(v4f*)&sB[0][r][c4] = v;
      }
      __syncthreads();
      compute_chunk(0);
      __syncthreads();
    }
#endif
  } else {
    // W[K,N]: gather 64 output columns, transpose into LDS (small GEMM only)
    for (int kc = 0; kc < K; kc += 64) {
      stage_A(kc);
#pragma unroll
      for (int i = 0; i < 8; ++i) {
        int idx = tid + i * 128;              // k = idx>>4, n4 = (idx&15)*4
        int k = idx >> 4, n4 = (idx & 15) << 2;
        v4f v = *(const v4f*)(W + (size_t)(kc + k) * N + n0 + n4);
        sB[0][n4 + 0][k] = v.x;
        sB[0][n4 + 1][k] = v.y;
        sB[0][n4 + 2][k] = v.z;
        sB[0][n4 + 3][k] = v.w;
      }
      __syncthreads();
      compute_chunk(0);
      __syncthreads();
    }
  }

#pragma unroll
  for (int j = 0; j < 4; ++j) {
    const int col = n0 + 16 * j + lm;
    const float bcol = bias ? bias[col] : 0.f;
#pragma unroll
    for (int jj = 0; jj < 8; ++jj) {        // C/D layout: row = jj + 8*hl
      int row = 16 * wv + jj + 8 * hl;
      float v = acc[j][jj] + bcol;
      if (EPI == 1) v = tanhf(v);
      C[(size_t)row * N + col] = v;
    }
  }
}

// ---------------------------------------------------------------------------
__global__ void embed_kernel(const int* __restrict__ ids,
                             const float* __restrict__ tab,
                             float* __restrict__ x) {
  int b = blockIdx.x;
  int id = ids[b];
  const v4f* src = (const v4f*)(tab + (size_t)id * E_SZ);
  v4f* dst = (v4f*)(x + (size_t)b * E_SZ);
  dst[threadIdx.x] = src[threadIdx.x];      // 128 threads * float4 = 512
}

// PyTorch GRU gate order [r, z, n]
__global__ void gru_pointwise(const float* __restrict__ gi,
                              const float* __restrict__ gh,
                              const float* __restrict__ hprev,
                              float* __restrict__ hnew) {
  int idx = blockIdx.x * blockDim.x + threadIdx.x;  // 64*1024
  int b = idx >> 10, c = idx & 1023;
  size_t base = (size_t)b * 3072;
  float r = sigmoidf_(gi[base + c] + gh[base + c]);
  float z = sigmoidf_(gi[base + 1024 + c] + gh[base + 1024 + c]);
  float n = tanhf(gi[base + 2048 + c] + r * gh[base + 2048 + c]);
  hnew[idx] = (1.f - z) * n + z * hprev[idx];
}

// scores -> mask -> softmax -> context, one block per batch row
__global__ __launch_bounds__(128) void attn_kernel(
    const float* __restrict__ enc, const float* __restrict__ qp,
    const unsigned char* __restrict__ mask, float* __restrict__ attn_out,
    float* __restrict__ ctx) {
  __shared__ float sw[L_SZ];
  __shared__ float red[L_SZ];
  int b = blockIdx.x, t = threadIdx.x;

  const float* row = enc + ((size_t)b * L_SZ + t) * H_SZ;
  const float* q = qp + (size_t)b * H_SZ;   // uniform across block -> broadcast
  float acc = 0.f;
  for (int k = 0; k < H_SZ; k += 4) {
    v4f e = *(const v4f*)(row + k);
    v4f qq = *(const v4f*)(q + k);
    acc += e.x * qq.x + e.y * qq.y + e.z * qq.z + e.w * qq.w;
  }
  float score = mask[(size_t)b * L_SZ + t] ? acc : -1e9f;

  red[t] = score;
  __syncthreads();
  for (int s = 64; s > 0; s >>= 1) {
    if (t < s) red[t] = fmaxf(red[t], red[t + s]);
    __syncthreads();
  }
  float m = red[0];
  __syncthreads();
  float e = __expf(score - m);
  red[t] = e;
  __syncthreads();
  for (int s = 64; s > 0; s >>= 1) {
    if (t < s) red[t] += red[t + s];
    __syncthreads();
  }
  float w = e / red[0];
  sw[t] = w;
  attn_out[(size_t)b * L_SZ + t] = w;
  __syncthreads();

  for (int h = t; h < H_SZ; h += 128) {     // coalesced along h
    float c = 0.f;
    for (int l = 0; l < L_SZ; ++l)
      c += sw[l] * enc[((size_t)b * L_SZ + l) * H_SZ + h];
    ctx[(size_t)b * H_SZ + h] = c;
  }
}

__global__ void concat_kernel(const float* __restrict__ h1,
                              const float* __restrict__ ctx,
                              float* __restrict__ cat) {
  int idx = blockIdx.x * blockDim.x + threadIdx.x;  // 64*2048
  int b = idx >> 11, c = idx & 2047;
  cat[idx] = (c < 1024) ? h1[(size_t)b * 1024 + c]
                        : ctx[(size_t)b * 1024 + (c - 1024)];
}

// ---------------------------------------------------------------------------
extern "C" void kernel_launch(void* const* d_in, const int* in_sizes, int n_in,
                              void* d_out, int out_size, void* d_ws,
                              size_t ws_size, hipStream_t stream) {
  const int* ids            = (const int*)d_in[0];
  const float* prev_h       = (const float*)d_in[1];   // [2,64,1024]
  const float* enc          = (const float*)d_in[2];   // [64,128,1024]
  const unsigned char* mask = (const unsigned char*)d_in[3];  // bool [64,128]
  const float* emb          = (const float*)d_in[4];
  const float* w_ih0 = (const float*)d_in[5];
  const float* w_hh0 = (const float*)d_in[6];
  const float* b_ih0 = (const float*)d_in[7];
  const float* b_hh0 = (const float*)d_in[8];
  const float* w_ih1 = (const float*)d_in[9];
  const float* w_hh1 = (const float*)d_in[10];
  const float* b_ih1 = (const float*)d_in[11];
  const float* b_hh1 = (const float*)d_in[12];
  const float* W_attn = (const float*)d_in[13];
  const float* W1 = (const float*)d_in[14];
  const float* b1 = (const float*)d_in[15];
  const float* W2 = (const float*)d_in[16];
  const float* b2 = (const float*)d_in[17];

  float* ws  = (float*)d_ws;
  float* x   = ws;               // 64*512
  float* gi  = ws + 32768;       // 64*3072
  float* gh  = ws + 229376;      // 64*3072
  float* qp  = ws + 425984;      // 64*1024
  float* ctx = ws + 491520;      // 64*1024
  float* cat = ws + 557056;      // 64*2048
  float* fc1 = ws + 688128;      // 64*1024

  float* out    = (float*)d_out;
  float* logits = out;                       // 64*32000
  float* h0     = out + 2048000;             // h[0]
  float* h1     = out + 2048000 + 65536;     // h[1]
  float* attn_w = out + 2048000 + 131072;    // 64*128

  embed_kernel<<<64, 128, 0, stream>>>(ids, emb, x);

  // GRU layer 0
  gemm64_wmma<0, false><<<48, 128, 0, stream>>>(x, w_ih0, b_ih0, gi, 512, 3072);
  gemm64_wmma<0, false><<<48, 128, 0, stream>>>(prev_h, w_hh0, b_hh0, gh, 1024, 3072);
  gru_pointwise<<<256, 256, 0, stream>>>(gi, gh, prev_h, h0);

  // GRU layer 1
  gemm64_wmma<0, false><<<48, 128, 0, stream>>>(h0, w_ih1, b_ih1, gi, 1024, 3072);
  gemm64_wmma<0, false><<<48, 128, 0, stream>>>(prev_h + 65536, w_hh1, b_hh1, gh, 1024, 3072);
  gru_pointwise<<<256, 256, 0, stream>>>(gi, gh, prev_h + 65536, h1);

  // Attention: qp = h1 @ W_attn  (algebraic rewrite of enc_proj einsum)
  gemm64_wmma<0, true><<<16, 128, 0, stream>>>(h1, W_attn, nullptr, qp, 1024, 1024);
  attn_kernel<<<64, 128, 0, stream>>>(enc, qp, mask, attn_w, ctx);

  // Output head
  concat_kernel<<<512, 256, 0, stream>>>(h1, ctx, cat);
  gemm64_wmma<1, false><<<16, 128, 0, stream>>>(cat, W1, b1, fc1, 2048, 1024);
  gemm64_wmma<0, false><<<500, 128, 0, stream>>>(fc1, W2, b2, logits, 1024, V_SZ);
}